// ALIF_85392539779873
// MI455X (gfx1250) — compile-verified
//
#include <hip/hip_runtime.h>
#include <cstdint>

// Problem constants (T,B,C,H,W) = (50,32,64,16,16)
#define T_STEPS 50
#define B_SZ 32
#define C_CH 64
#define HW 256                              // H*W
#define N_SPATIAL (B_SZ * C_CH * HW)        // 524288 elements per timestep
#define NE_F 409600.0f                      // T*B*HW elements per channel

// Stats pass decomposition: 1600 HW-planes per channel, 25 splits of 64 planes,
// 4 planes per block-iteration (256 threads x float4 = 4 planes of 256 floats).
#define S_SPLIT 25
#define PLANES_PER_BLK 64

typedef float vfloat4 __attribute__((ext_vector_type(4)));

// ---------------------------------------------------------------------------
// Pass 1: per-channel partial sum / sum-of-squares.
// grid = (S_SPLIT, C_CH), block = 256. Each lane streams float4s with a fixed
// 256KB pointer stride -> B128 loads, no per-iteration address rebuild.
// Regular (RT) loads intentionally pull x (105MB) into the 192MB L2 so the
// scan pass re-reads it from L2 instead of HBM.
// ---------------------------------------------------------------------------
__global__ void alif_stats(const float* __restrict__ x,
                           float* __restrict__ psum,
                           float* __restrict__ psq) {
  const int tid   = threadIdx.x;
  const int sblk  = blockIdx.x;          // 0..24
  const int c     = blockIdx.y;          // channel
  const int lane4 = tid & 63;            // float4 slot within a plane
  const int psub  = tid >> 6;            // plane sub-index 0..3
  const int p0    = sblk * PLANES_PER_BLK + psub;

  const vfloat4* ptr =
      (const vfloat4*)(x + ((size_t)p0 * C_CH + c) * HW) + lane4;
  const size_t stride4 = (size_t)4 * C_CH * HW / 4;   // 4 planes, in float4s

  float sum = 0.0f, sq = 0.0f;
#pragma unroll 4
  for (int it = 0; it < PLANES_PER_BLK / 4; ++it) {
    vfloat4 v = *ptr;
    ptr += stride4;
    sum += (v.x + v.y) + (v.z + v.w);
    sq = fmaf(v.x, v.x, sq);
    sq = fmaf(v.y, v.y, sq);
    sq = fmaf(v.z, v.z, sq);
    sq = fmaf(v.w, v.w, sq);
  }

  __shared__ float ssum[256];
  __shared__ float ssq[256];
  ssum[tid] = sum;
  ssq[tid]  = sq;
  __syncthreads();
  for (int off = 128; off > 0; off >>= 1) {
    if (tid < off) {
      ssum[tid] += ssum[tid + off];
      ssq[tid]  += ssq[tid + off];
    }
    __syncthreads();
  }
  if (tid == 0) {
    psum[c * S_SPLIT + sblk] = ssum[0];
    psq [c * S_SPLIT + sblk] = ssq[0];
  }
}

// ---------------------------------------------------------------------------
// Pass 1b: fold partials -> per-channel affine (scale, shift).
// xn = x*scale + shift with scale = w*rsqrt(var+eps), shift = b - mean*scale.
// ---------------------------------------------------------------------------
__global__ void alif_finalize(const float* __restrict__ psum,
                              const float* __restrict__ psq,
                              const float* __restrict__ w,
                              const float* __restrict__ b,
                              float* __restrict__ scale,
                              float* __restrict__ shift) {
  const int c = threadIdx.x;   // 0..63
  float s = 0.0f, q = 0.0f;
  for (int k = 0; k < S_SPLIT; ++k) {
    s += psum[c * S_SPLIT + k];
    q += psq [c * S_SPLIT + k];
  }
  const float mean = s / NE_F;
  const float var  = q / NE_F - mean * mean;    // population variance (ddof=0)
  const float sc   = w[c] * rsqrtf(var + 1.0e-4f);
  scale[c] = sc;
  shift[c] = fmaf(-mean, sc, b[c]);
}

// ---------------------------------------------------------------------------
// Pass 2: ALIF temporal scan. Each thread owns 4 consecutive spatial elements
// (same channel) and 4 independent (mem, adapt) recurrences. Timestep slices
// are staged LDS-side with a double-buffered global_load_async_to_lds_b128
// pipeline (ASYNCcnt): slice t+1 streams in while slice t is computed.
// Ordering:
//   s_wait_dscnt 0   -> all prior LDS reads done before the async load that
//                       overwrites their buffer is issued (kills the WAR race)
//   issue async t+1
//   s_wait_asynccnt 1-> slice t has landed in LDS (loads complete in order)
// Spike output uses non-temporal B128 stores so x stays L2-resident.
// ---------------------------------------------------------------------------
__global__ void alif_scan(const float* __restrict__ x,
                          const float* __restrict__ scale,
                          const float* __restrict__ shift,
                          float* __restrict__ out) {
  const int tid = threadIdx.x;
  const int g   = blockIdx.x * 256 + tid;        // float4 index
  const int i4  = g * 4;                         // element index
  const int c   = (i4 >> 8) & (C_CH - 1);        // channel = (i/HW)%C
  const float sc = scale[c];
  const float sh = shift[c];

  __shared__ float buf[2][1024];                 // 2 x 4KB staging buffers
  // Low 32 bits of the generic shared-space address == LDS byte offset.
  const unsigned ldsA = (unsigned)(unsigned long long)(&buf[0][tid * 4]);
  const unsigned ldsB = (unsigned)(unsigned long long)(&buf[1][tid * 4]);

  unsigned voff = (unsigned)i4 * 4u;             // byte offset of slice t=0
  const unsigned vstride = (unsigned)N_SPATIAL * 4u;

  // Prologue: stage t = 0.
  asm volatile("global_load_async_to_lds_b128 %0, %1, %2"
               :: "v"(ldsA), "v"(voff), "s"(x) : "memory");

  float mem[4]   = {0.0f, 0.0f, 0.0f, 0.0f};
  float adapt[4] = {0.0f, 0.0f, 0.0f, 0.0f};

  for (int t = 0; t < T_STEPS; ++t) {
    if (t + 1 < T_STEPS) {
      voff += vstride;
      unsigned lds = ((t + 1) & 1) ? ldsB : ldsA;
      asm volatile("s_wait_dscnt 0x0" ::: "memory");
      asm volatile("global_load_async_to_lds_b128 %0, %1, %2"
                   :: "v"(lds), "v"(voff), "s"(x) : "memory");
      asm volatile("s_wait_asynccnt 0x1" ::: "memory");
    } else {
      asm volatile("s_wait_asynccnt 0x0" ::: "memory");
    }

    const vfloat4 v = *(const vfloat4*)((t & 1) ? &buf[1][tid * 4]
                                                : &buf[0][tid * 4]);
    vfloat4 spike;
#pragma unroll
    for (int k = 0; k < 4; ++k) {
      const float xn = fmaf(v[k], sc, sh);             // batchnorm affine
      mem[k] = fmaf(0.9f, mem[k], xn);                 // mem = BETA*mem + xn
      const float s = (mem[k] - (1.0f + adapt[k])) > 0.0f ? 1.0f : 0.0f;
      spike[k] = s;
      mem[k]   = (s > 0.0f) ? 0.0f : mem[k];           // reset on spike
      adapt[k] = fmaf(0.96f, adapt[k], 0.5f * s);      // adaptation dynamics
    }
    __builtin_nontemporal_store(
        spike, (vfloat4*)(out + (size_t)t * N_SPATIAL + i4));
  }
}

// ---------------------------------------------------------------------------
// Launcher. Inputs: d_in[0]=x (T,B,C,H,W) f32, d_in[1]=bn_weight (C),
// d_in[2]=bn_bias (C). Output: spikes (T,B,C,H,W) f32.
// Workspace layout (floats): psum[C*S] | psq[C*S] | scale[C] | shift[C]
// ---------------------------------------------------------------------------
extern "C" void kernel_launch(void* const* d_in, const int* in_sizes, int n_in,
                              void* d_out, int out_size, void* d_ws, size_t ws_size,
                              hipStream_t stream) {
  (void)in_sizes; (void)n_in; (void)out_size; (void)ws_size;
  const float* x  = (const float*)d_in[0];
  const float* bw = (const float*)d_in[1];
  const float* bb = (const float*)d_in[2];
  float* out = (float*)d_out;

  float* psum  = (float*)d_ws;                 // C*S = 1600 floats
  float* psq   = psum + C_CH * S_SPLIT;        // 1600 floats
  float* scale = psq  + C_CH * S_SPLIT;        // 64 floats
  float* shift = scale + C_CH;                 // 64 floats

  alif_stats   <<<dim3(S_SPLIT, C_CH), 256, 0, stream>>>(x, psum, psq);
  alif_finalize<<<1, C_CH, 0, stream>>>(psum, psq, bw, bb, scale, shift);
  alif_scan    <<<N_SPATIAL / 1024, 256, 0, stream>>>(x, scale, shift, out);
}